// BernNetHead_75780402970870
// MI455X (gfx1250) — compile-verified
//
#include <hip/hip_runtime.h>

// ---------------------------------------------------------------------------
// BernNet head, algebraically collapsed:
//   logits = ( (1/N) * (g(L)^T 1)^T  relu(X W1 + b1) ) W2 + b2
// Sizes: B=8, N=2048, F0=128, HID=64, OUT=16, K=10
// ---------------------------------------------------------------------------

#define NN   2048
#define F0   128
#define HID  64
#define OUTD 16
#define BB   8
#define KK   10

typedef __attribute__((ext_vector_type(2))) float v2f;
typedef __attribute__((ext_vector_type(8))) float v8f;

// ---------------------------------------------------------------------------
// Step 0: ys[0] = ones, r = theta[K] * C(K,K) = theta[10]
// ---------------------------------------------------------------------------
__global__ void bern_init(float* __restrict__ ones_v, float* __restrict__ r,
                          const float* __restrict__ theta) {
    int w = blockIdx.x * blockDim.x + threadIdx.x;
    if (w < NN) {
        ones_v[w] = 1.0f;
        r[w]      = theta[KK];
    }
}

// ---------------------------------------------------------------------------
// out[w] = sgn * sum_v L[v,w] * x[v]  +  alpha * z[w]
//   alpha = binom * (tidx >= 0 ? theta[tidx] : 1)
// Used for both:
//   ys step:     out = x - L^T x          (sgn=-1, alpha=1, z=x)
//   Horner step: out = L^T r + a_i * ys_j (sgn=+1, alpha=theta[i]*C(K,i))
// Column-parallel: consecutive lanes read consecutive columns -> coalesced.
// L (16 MB) stays resident in the 192 MB L2 across all 20 sweeps.
// ---------------------------------------------------------------------------
__global__ void bern_matvec_axpy(const float* __restrict__ L,
                                 const float* __restrict__ x,
                                 const float* __restrict__ z,
                                 const float* __restrict__ theta,
                                 int tidx, float binom, float sgn,
                                 float* __restrict__ out) {
    int w = blockIdx.x * blockDim.x + threadIdx.x;
    if (w >= NN) return;
    float alpha = binom;
    if (tidx >= 0) alpha *= theta[tidx];
    float acc = 0.0f;
#pragma unroll 8
    for (int v = 0; v < NN; ++v) {
        acc = fmaf(L[(size_t)v * NN + w], x[v], acc);
    }
    out[w] = sgn * acc + alpha * z[w];
}

// ---------------------------------------------------------------------------
// Fused: Hf = relu(X W1 + b1) via V_WMMA_F32_16X16X4_F32 (exact fp32),
// then per-tile weighted column sums  partial[tile][f] = sum_m u[w(m)]*Hf[m,f].
// One wave = one 16x16 C tile. 4 waves/block cover HID=64. 1024 blocks cover
// the 16384 flattened (B*N) rows. EXEC is all-ones everywhere WMMA executes.
//
// fp32 A 16x4 layout (ISA 7.12.2): lanes 0-15 -> M=lane, {K0,K1} in v[0],v[1];
// lanes 16-31 -> {K2,K3}. B 4x16: VGPR v, lane group g: K = v + 2g, N=lane%16.
// C/D: VGPR r, lanes 0-15 -> M=r, lanes 16-31 -> M=r+8, N=lane%16.
// ---------------------------------------------------------------------------
__global__ void gemm_relu_usum(const float* __restrict__ X,
                               const float* __restrict__ W1,
                               const float* __restrict__ b1,
                               const float* __restrict__ u,
                               float* __restrict__ partial) {
    const int tile  = blockIdx.x;          // 16-row strip of flattened rows
    const int wave  = threadIdx.x >> 5;    // 0..3 -> output col base
    const int lane  = threadIdx.x & 31;
    const int m0    = tile * 16;
    const int c0    = wave * 16;
    const int n     = lane & 15;           // col within tile / row M for A
    const int khalf = (lane >> 4) * 2;     // 0 or 2

    const float* __restrict__ xrow = X + (size_t)(m0 + n) * F0;

    v8f acc = {};
#pragma unroll
    for (int k = 0; k < F0; k += 4) {
        v2f a, b;
        a.x = xrow[k + khalf];
        a.y = xrow[k + khalf + 1];
        b.x = W1[(size_t)(k + khalf + 0) * HID + c0 + n];
        b.y = W1[(size_t)(k + khalf + 1) * HID + c0 + n];
        acc = __builtin_amdgcn_wmma_f32_16x16x4_f32(
            /*neg_a=*/false, a, /*neg_b=*/false, b,
            /*c_mod=*/(short)0, acc, /*reuse_a=*/false, /*reuse_b=*/false);
    }

    // Epilogue: bias + relu + u-weight, reduce the 16 rows of this tile.
    const float bias   = b1[c0 + n];
    const float invN   = 1.0f / (float)NN;
    float colsum = 0.0f;
#pragma unroll
    for (int r = 0; r < 8; ++r) {
        int m    = r + ((lane >> 4) << 3);       // row within tile
        int grow = m0 + m;                       // flattened (b*N + w)
        int w    = grow & (NN - 1);
        float val = acc[r] + bias;
        val = val > 0.0f ? val : 0.0f;
        colsum = fmaf(val, u[w] * invN, colsum);
    }
    // lanes l and l^16 hold the two 8-row halves of the same column
    colsum += __shfl_xor(colsum, 16, 32);
    if (lane < 16) partial[(size_t)tile * HID + c0 + n] = colsum;
}

// ---------------------------------------------------------------------------
// Reduce 128 tiles/batch -> Hg[8][64], then logits = Hg @ W2 + b2.
// One block, 512 threads; deterministic fixed-order sums.
// ---------------------------------------------------------------------------
__global__ void bern_finalize(const float* __restrict__ partial,
                              const float* __restrict__ W2,
                              const float* __restrict__ b2,
                              float* __restrict__ out) {
    __shared__ float Hg[BB * HID];
    const int t = threadIdx.x;             // 0..511
    const int b = t >> 6;                  // batch
    const int f = t & (HID - 1);           // feature
    float s = 0.0f;
    const int tiles_per_batch = NN / 16;   // 128
#pragma unroll 4
    for (int tb = 0; tb < tiles_per_batch; ++tb)
        s += partial[(size_t)(b * tiles_per_batch + tb) * HID + f];
    Hg[t] = s;
    __syncthreads();
    if (t < BB * OUTD) {
        const int bb = t >> 4;
        const int o  = t & (OUTD - 1);
        float accv = b2[o];
#pragma unroll
        for (int ff = 0; ff < HID; ++ff)
            accv = fmaf(Hg[bb * HID + ff], W2[ff * OUTD + o], accv);
        out[bb * OUTD + o] = accv;
    }
}

// ---------------------------------------------------------------------------
// Host-side orchestration (graph-capture safe: launches only).
// ---------------------------------------------------------------------------
extern "C" void kernel_launch(void* const* d_in, const int* in_sizes, int n_in,
                              void* d_out, int out_size, void* d_ws, size_t ws_size,
                              hipStream_t stream) {
    const float* X     = (const float*)d_in[0];  // (8,2048,128)
    const float* L     = (const float*)d_in[1];  // (2048,2048)
    const float* W1    = (const float*)d_in[2];  // (128,64)
    const float* b1    = (const float*)d_in[3];  // (64,)
    const float* W2    = (const float*)d_in[4];  // (64,16)
    const float* b2    = (const float*)d_in[5];  // (16,)
    const float* theta = (const float*)d_in[6];  // (11,)
    float* out = (float*)d_out;                  // (8,16)

    // Workspace layout (floats)
    float* ws      = (float*)d_ws;
    float* ys      = ws;                       // (K+1) * NN
    float* rA      = ys + (KK + 1) * NN;       // NN
    float* rB      = rA + NN;                  // NN
    float* partial = rB + NN;                  // 1024 * 64

    const float binom[KK + 1] = {1.f, 10.f, 45.f, 120.f, 210.f, 252.f,
                                 210.f, 120.f, 45.f, 10.f, 1.f};

    const dim3 vb(256), vg((NN + 255) / 256);  // 8 blocks of 256 for vector ops

    // u-chain: ys[k] = (I - L^T)^k 1
    bern_init<<<vg, vb, 0, stream>>>(ys, rA, theta);
    for (int k = 1; k <= KK; ++k) {
        const float* prev = ys + (size_t)(k - 1) * NN;
        bern_matvec_axpy<<<vg, vb, 0, stream>>>(
            L, prev, prev, theta, /*tidx=*/-1, /*binom=*/1.0f, /*sgn=*/-1.0f,
            ys + (size_t)k * NN);
    }
    // Horner: r <- L^T r + theta[i]*C(K,i) * ys[K-i], i = K-1..0
    float* rin = rA;
    float* rot = rB;
    for (int i = KK - 1; i >= 0; --i) {
        bern_matvec_axpy<<<vg, vb, 0, stream>>>(
            L, rin, ys + (size_t)(KK - i) * NN, theta, /*tidx=*/i,
            binom[i], /*sgn=*/+1.0f, rot);
        float* tmp = rin; rin = rot; rot = tmp;
    }
    // After 10 swaps, result is back in rA (== rin).

    // WMMA GEMM + fused relu/bias/u-weighted tile reduction
    const int row_tiles = (BB * NN) / 16;      // 1024
    gemm_relu_usum<<<dim3(row_tiles), dim3(128), 0, stream>>>(
        X, W1, b1, rin, partial);

    // Final reduction + tiny output GEMM
    bern_finalize<<<dim3(1), dim3(512), 0, stream>>>(partial, W2, b2, out);
}